// FactoredHmmLm_77214922047934
// MI455X (gfx1250) — compile-verified
//
#include <hip/hip_runtime.h>

#define HD   256
#define SW   64
#define KCL  128
#define CST  8192
#define VSZ  10000
#define NBAT 16
#define TLEN 256

typedef __attribute__((ext_vector_type(2))) float v2f;
typedef __attribute__((ext_vector_type(8))) float v8f;

// ---- CDNA5 f32 WMMA: D(16x16) = A(16x4) * B(4x16) + C ----
// A(m,k): lane = m + 16*(k>>1), reg = k&1.  B(k,n): lane = n + 16*(k>>1), reg = k&1.
// C/D(g): m = g + 8*(lane>>4), n = lane&15.
__device__ __forceinline__ v8f wmma4(v2f a, v2f b, v8f c) {
  return __builtin_amdgcn_wmma_f32_16x16x4_f32(false, a, false, b, (short)0, c, false, false);
}

// ---- numerically safe streaming logsumexp merge ----
__device__ __forceinline__ void lse_merge(float& M, float& S, float m2, float s2) {
  float nm = fmaxf(M, m2);
  if (nm == -INFINITY) { M = nm; S = 0.0f; return; }
  S = S * expf(M - nm) + s2 * expf(m2 - nm);
  M = nm;
}
__device__ __forceinline__ void lse_acc(float& M, float& S, float v) { lse_merge(M, S, v, 1.0f); }

// ================= embeddings: out[c*SW+s][h] = ec[c][h] + es[s][h] =================
__global__ __launch_bounds__(256) void k_emb(const float* __restrict__ ec,
                                             const float* __restrict__ es,
                                             float* __restrict__ out) {
  int idx = blockIdx.x * 256 + threadIdx.x;       // over CST*HD
  int h  = idx & (HD - 1);
  int st = idx >> 8;
  int s  = st & (SW - 1);
  int c  = st >> 6;
  out[idx] = ec[c * HD + h] + es[s * HD + h];
}

// ================= generic OUT = relu?(A @ B^T + bias + skip) via f32 WMMA =================
// one wave = one 16x16 tile; M,N multiples of 16; K multiple of 4.
__global__ __launch_bounds__(256) void k_gemm(const float* __restrict__ A,
                                              const float* __restrict__ B,
                                              const float* __restrict__ bias,
                                              const float* __restrict__ skip,
                                              float* __restrict__ OUT,
                                              int M, int Ncols, int Kdim, int relu) {
  int w = threadIdx.x >> 5, lane = threadIdx.x & 31;
  int tile = blockIdx.x * 8 + w;
  int ntN = Ncols >> 4;
  int mt = tile / ntN, nt = tile - mt * ntN;
  if (mt * 16 >= M) return;
  int r = lane & 15, kh = lane >> 4;
  const float* arow = A + (size_t)(mt * 16 + r) * Kdim + 2 * kh;
  const float* brow = B + (size_t)(nt * 16 + r) * Kdim + 2 * kh;
  v8f acc = {0.f,0.f,0.f,0.f,0.f,0.f,0.f,0.f};
  for (int k0 = 0; k0 < Kdim; k0 += 4)
    acc = wmma4(*(const v2f*)(arow + k0), *(const v2f*)(brow + k0), acc);
  int n = nt * 16 + r;
  float bv = bias ? bias[n] : 0.0f;
#pragma unroll
  for (int g = 0; g < 8; ++g) {
    int m = mt * 16 + g + 8 * kh;
    float v = acc[g] + bv;
    if (skip) v += skip[(size_t)m * Ncols + n];
    if (relu) v = fmaxf(v, 0.0f);
    OUT[(size_t)m * Ncols + n] = v;
  }
}

// ================= start scores: scores[i] = dot(sh[i], w) + b =================
__global__ __launch_bounds__(256) void k_start_scores(const float* __restrict__ sh,
                                                      const float* __restrict__ wv,
                                                      const float* __restrict__ bv,
                                                      float* __restrict__ scores) {
  int row  = (blockIdx.x * 256 + threadIdx.x) >> 2;
  int part = threadIdx.x & 3;
  const float* xr = sh + (size_t)row * HD + part * 64;
  const float* wr = wv + part * 64;
  float s = 0.0f;
#pragma unroll 8
  for (int j = 0; j < 64; ++j) s += xr[j] * wr[j];
  s += __shfl_xor(s, 1, 32);
  s += __shfl_xor(s, 2, 32);
  if (part == 0) scores[row] = s + bv[0];
}

// ================= single-block LSE over a vector -> scalar =================
__global__ __launch_bounds__(256) void k_vec_lse(const float* __restrict__ v, int n,
                                                 float* __restrict__ out) {
  __shared__ float sm_[256], ss_[256];
  int tid = threadIdx.x;
  float M = -INFINITY, S = 0.0f;
  for (int i = tid; i < n; i += 256) lse_acc(M, S, v[i]);
  sm_[tid] = M; ss_[tid] = S;
  __syncthreads();
  for (int off = 128; off; off >>= 1) {
    if (tid < off) {
      float m = sm_[tid], s = ss_[tid];
      lse_merge(m, s, sm_[tid + off], ss_[tid + off]);
      sm_[tid] = m; ss_[tid] = s;
    }
    __syncthreads();
  }
  if (tid == 0) out[0] = sm_[0] + logf(ss_[0]);
}

// ================= rowLSE[i] = LSE_j x[i].nse[j], 64 rows per block =================
// dyn LDS: 64*256 floats (X block), reused afterwards for cross-wave partials.
__global__ __launch_bounds__(256) void k_row_lse(const float* __restrict__ X,
                                                 const float* __restrict__ Y,
                                                 float* __restrict__ rowlse) {
  extern __shared__ float smem[];
  int tid = threadIdx.x;
  int row0 = blockIdx.x * 64;
  for (int i = tid; i < 64 * HD / 4; i += 256)
    ((float4*)smem)[i] = ((const float4*)(X + (size_t)row0 * HD))[i];
  __syncthreads();
  int w = tid >> 5, lane = tid & 31, r = lane & 15, kh = lane >> 4;
  float runM[4][8], runS[4][8];
#pragma unroll
  for (int rt = 0; rt < 4; ++rt)
#pragma unroll
    for (int g = 0; g < 8; ++g) { runM[rt][g] = -INFINITY; runS[rt][g] = 0.0f; }
  const int tilesPerWave = (CST / 16) / 8;   // 64
  for (int ct = 0; ct < tilesPerWave; ++ct) {
    int col0 = (w * tilesPerWave + ct) * 16;
    const float* brow = Y + (size_t)(col0 + r) * HD + 2 * kh;
    v8f acc[4];
#pragma unroll
    for (int rt = 0; rt < 4; ++rt) acc[rt] = (v8f){0.f,0.f,0.f,0.f,0.f,0.f,0.f,0.f};
    for (int k0 = 0; k0 < HD; k0 += 4) {
      v2f b = *(const v2f*)(brow + k0);
#pragma unroll
      for (int rt = 0; rt < 4; ++rt) {
        v2f a = *(const v2f*)(smem + (rt * 16 + r) * HD + 2 * kh + k0);
        acc[rt] = wmma4(a, b, acc[rt]);
      }
    }
#pragma unroll
    for (int rt = 0; rt < 4; ++rt)
#pragma unroll
      for (int g = 0; g < 8; ++g) lse_acc(runM[rt][g], runS[rt][g], acc[rt][g]);
  }
  // reduce over 16 lanes of each half-wave (same row, different columns)
#pragma unroll
  for (int rt = 0; rt < 4; ++rt)
#pragma unroll
    for (int g = 0; g < 8; ++g) {
      float M = runM[rt][g], S = runS[rt][g];
      for (int d = 1; d < 16; d <<= 1) {
        float m2 = __shfl_xor(M, d, 32), s2 = __shfl_xor(S, d, 32);
        lse_merge(M, S, m2, s2);
      }
      runM[rt][g] = M; runS[rt][g] = S;
    }
  __syncthreads();                 // done using smem as X
  float* pm = smem;                // [8][64]
  float* ps = smem + 8 * 64;
  if (r == 0) {
#pragma unroll
    for (int rt = 0; rt < 4; ++rt)
#pragma unroll
      for (int g = 0; g < 8; ++g) {
        int m = rt * 16 + g + 8 * kh;
        pm[w * 64 + m] = runM[rt][g];
        ps[w * 64 + m] = runS[rt][g];
      }
  }
  __syncthreads();
  if (tid < 64) {
    float M = pm[tid], S = ps[tid];
    for (int ww = 1; ww < 8; ++ww) lse_merge(M, S, pm[ww * 64 + tid], ps[ww * 64 + tid]);
    rowlse[row0 + tid] = M + logf(S);
  }
}

// ================= deterministic word->cluster bucketing =================
__global__ void k_offsets(const int* __restrict__ w2c, int* __restrict__ off) {
  __shared__ int cnt[KCL];
  int c = threadIdx.x;            // 128 threads
  int n = 0;
  for (int v = 0; v < VSZ; ++v) n += (w2c[v] == c);
  cnt[c] = n;
  __syncthreads();
  if (c == 0) {
    int acc = 0;
    for (int i = 0; i < KCL; ++i) { off[i] = acc; acc += cnt[i]; }
    off[KCL] = acc;
  }
}
__global__ void k_fill(const int* __restrict__ w2c, const int* __restrict__ off,
                       int* __restrict__ list) {
  int c = blockIdx.x, lane = threadIdx.x;   // 32 threads (one wave)
  int idx = off[c];
  for (int v0 = 0; v0 < VSZ; v0 += 32) {
    int v = v0 + lane;
    bool p = (v < VSZ) && (w2c[v] == c);
    unsigned mm = (unsigned)__ballot(p);
    if (p) list[idx + __popc(mm & ((1u << lane) - 1u))] = v;
    idx += __popc(mm);
  }
}

// ================= Z[state] = LSE over words in cluster of (ph[state].w_v + b_v) =================
__global__ __launch_bounds__(256) void k_Z(const float* __restrict__ ph,
                                           const float* __restrict__ W,
                                           const float* __restrict__ Wb,
                                           const int* __restrict__ off,
                                           const int* __restrict__ list,
                                           float* __restrict__ Z) {
  extern __shared__ float smem[];
  int c = blockIdx.x, tid = threadIdx.x;
  int base = off[c], cnt = off[c + 1] - base;
  if (cnt == 0) { if (tid < 64) Z[c * 64 + tid] = -INFINITY; return; }
  for (int i = tid; i < 64 * HD / 4; i += 256)
    ((float4*)smem)[i] = ((const float4*)(ph + (size_t)c * 64 * HD))[i];
  __syncthreads();
  int w = tid >> 5, lane = tid & 31, r = lane & 15, kh = lane >> 4;
  float runM[4][8], runS[4][8];
#pragma unroll
  for (int rt = 0; rt < 4; ++rt)
#pragma unroll
    for (int g = 0; g < 8; ++g) { runM[rt][g] = -INFINITY; runS[rt][g] = 0.0f; }
  int ntiles = (cnt + 15) >> 4;
  for (int ct = w; ct < ntiles; ct += 8) {
    int colr = ct * 16 + r;
    int ci = colr < cnt ? colr : cnt - 1;
    int v = list[base + ci];
    bool valid = colr < cnt;
    const float* brow = W + (size_t)v * HD + 2 * kh;
    float bias = Wb[v];
    v8f acc[4];
#pragma unroll
    for (int rt = 0; rt < 4; ++rt) acc[rt] = (v8f){0.f,0.f,0.f,0.f,0.f,0.f,0.f,0.f};
    for (int k0 = 0; k0 < HD; k0 += 4) {
      v2f b = *(const v2f*)(brow + k0);
#pragma unroll
      for (int rt = 0; rt < 4; ++rt) {
        v2f a = *(const v2f*)(smem + (rt * 16 + r) * HD + 2 * kh + k0);
        acc[rt] = wmma4(a, b, acc[rt]);
      }
    }
#pragma unroll
    for (int rt = 0; rt < 4; ++rt)
#pragma unroll
      for (int g = 0; g < 8; ++g) {
        float val = valid ? (acc[rt][g] + bias) : -INFINITY;
        lse_acc(runM[rt][g], runS[rt][g], val);
      }
  }
#pragma unroll
  for (int rt = 0; rt < 4; ++rt)
#pragma unroll
    for (int g = 0; g < 8; ++g) {
      float M = runM[rt][g], S = runS[rt][g];
      for (int d = 1; d < 16; d <<= 1) {
        float m2 = __shfl_xor(M, d, 32), s2 = __shfl_xor(S, d, 32);
        lse_merge(M, S, m2, s2);
      }
      runM[rt][g] = M; runS[rt][g] = S;
    }
  __syncthreads();
  float* pm = smem;
  float* ps = smem + 8 * 64;
  if (r == 0) {
#pragma unroll
    for (int rt = 0; rt < 4; ++rt)
#pragma unroll
      for (int g = 0; g < 8; ++g) {
        int m = rt * 16 + g + 8 * kh;
        pm[w * 64 + m] = runM[rt][g];
        ps[w * 64 + m] = runS[rt][g];
      }
  }
  __syncthreads();
  if (tid < 64) {
    float M = pm[tid], S = ps[tid];
    for (int ww = 1; ww < 8; ++ww) lse_merge(M, S, pm[ww * 64 + tid], ps[ww * 64 + tid]);
    Z[c * 64 + tid] = M + logf(S);
  }
}

// ================= emit[n,t,s] = ph[state].w_word + b_word - Z[state] =================
__global__ __launch_bounds__(256) void k_emit(const float* __restrict__ ph,
                                              const float* __restrict__ W,
                                              const float* __restrict__ Wb,
                                              const float* __restrict__ Z,
                                              const int* __restrict__ text,
                                              const int* __restrict__ w2c,
                                              float* __restrict__ emit) {
  __shared__ float sw[HD];
  int nt = blockIdx.x;             // n*T + t
  int word = text[nt];
  int c = w2c[word];
  int tid = threadIdx.x;
  sw[tid] = W[(size_t)word * HD + tid];
  __syncthreads();
  int s = tid >> 2, part = tid & 3;
  const float* pr = ph + (size_t)(c * 64 + s) * HD + part * 64;
  const float* wr = sw + part * 64;
  float acc = 0.0f;
#pragma unroll 8
  for (int j = 0; j < 64; ++j) acc += pr[j] * wr[j];
  acc += __shfl_xor(acc, 1, 32);
  acc += __shfl_xor(acc, 2, 32);
  if (part == 0) emit[nt * 64 + s] = acc + Wb[word] - Z[c * 64 + s];
}

// ================= forward recursion: one block per batch element =================
// Next step's x/nse blocks (two contiguous 64KB regions, known one step ahead)
// are prefetched via global_prefetch_b8 to overlap cache fill with the WMMA loop.
__global__ __launch_bounds__(256) void k_rec(const int* __restrict__ text,
                                             const int* __restrict__ w2c,
                                             const float* __restrict__ X,
                                             const float* __restrict__ Y,
                                             const float* __restrict__ rowlse,
                                             const float* __restrict__ scores,
                                             const float* __restrict__ slse,
                                             const float* __restrict__ emit,
                                             float* __restrict__ out) {
  __shared__ float s_alpha[64], s_arow[64], s_pm[128], s_ps[128];
  int n = blockIdx.x, tid = threadIdx.x;
  int w = tid >> 5, lane = tid & 31, r = lane & 15, kh = lane >> 4;
  const int* tx = text + n * TLEN;
  if (tid < 64) {
    int c0 = w2c[tx[0]];
    s_alpha[tid] = scores[c0 * 64 + tid] - slse[0] + emit[(n * TLEN + 0) * 64 + tid];
  }
  __syncthreads();
  int nt = w & 3;
  int mt0 = (w >> 2) * 2;
  for (int t = 1; t < TLEN; ++t) {
    int cp = w2c[tx[t - 1]], cc = w2c[tx[t]];
    // prefetch operand blocks for step t+1 (x-block cluster for t+1 is cc)
    if (t + 1 < TLEN) {
      int ccn = w2c[tx[t + 1]];
      const char* px = (const char*)(X + (size_t)cc  * 64 * HD) + tid * 256;
      const char* py = (const char*)(Y + (size_t)ccn * 64 * HD) + tid * 256;
      __builtin_prefetch(px, 0, 3);
      __builtin_prefetch(px + 128, 0, 3);
      __builtin_prefetch(py, 0, 3);
      __builtin_prefetch(py + 128, 0, 3);
    }
    if (tid < 64) s_arow[tid] = s_alpha[tid] - rowlse[cp * 64 + tid];
    __syncthreads();
    const float* b  = Y + (size_t)(cc * 64 + nt * 16 + r) * HD + 2 * kh;
    const float* a0 = X + (size_t)(cp * 64 + mt0 * 16 + r) * HD + 2 * kh;
    const float* a1 = a0 + 16 * HD;
    v8f acc0 = (v8f){0.f,0.f,0.f,0.f,0.f,0.f,0.f,0.f};
    v8f acc1 = (v8f){0.f,0.f,0.f,0.f,0.f,0.f,0.f,0.f};
    for (int k0 = 0; k0 < HD; k0 += 4) {
      v2f bv = *(const v2f*)(b + k0);
      acc0 = wmma4(*(const v2f*)(a0 + k0), bv, acc0);
      acc1 = wmma4(*(const v2f*)(a1 + k0), bv, acc1);
    }
    float M = -INFINITY, S = 0.0f;
#pragma unroll
    for (int g = 0; g < 8; ++g) {
      int i0 = mt0 * 16 + g + 8 * kh;
      lse_acc(M, S, acc0[g] + s_arow[i0]);
      lse_acc(M, S, acc1[g] + s_arow[i0 + 16]);
    }
    { // lanes l and l^16 hold the same column j = nt*16 + r, different i halves
      float m2 = __shfl_xor(M, 16, 32), s2 = __shfl_xor(S, 16, 32);
      lse_merge(M, S, m2, s2);
    }
    if (lane < 16) { s_pm[w * 16 + lane] = M; s_ps[w * 16 + lane] = S; }
    __syncthreads();
    if (tid < 64) {
      int jn = tid >> 4, jc = tid & 15;
      float M2 = s_pm[jn * 16 + jc], S2 = s_ps[jn * 16 + jc];
      lse_merge(M2, S2, s_pm[(jn + 4) * 16 + jc], s_ps[(jn + 4) * 16 + jc]);
      s_alpha[tid] = M2 + logf(S2) + emit[(n * TLEN + t) * 64 + tid];
    }
    __syncthreads();
  }
  if (tid == 0) {
    float M = -INFINITY, S = 0.0f;
    for (int j = 0; j < 64; ++j) lse_acc(M, S, s_alpha[j]);
    out[n] = M + logf(S);
  }
}

// ================= orchestration =================
extern "C" void kernel_launch(void* const* d_in, const int* in_sizes, int n_in,
                              void* d_out, int out_size, void* d_ws, size_t ws_size,
                              hipStream_t stream) {
  const int*   text  = (const int*)d_in[0];
  const int*   w2c   = (const int*)d_in[1];
  const float* emb_c[4] = {(const float*)d_in[2], (const float*)d_in[4],
                           (const float*)d_in[6], (const float*)d_in[8]};
  const float* emb_s[4] = {(const float*)d_in[3], (const float*)d_in[5],
                           (const float*)d_in[7], (const float*)d_in[9]};
  // residual weights: start(10..13) trans(14..17) term(18..21)
  const float* rw1[3] = {(const float*)d_in[10], (const float*)d_in[14], (const float*)d_in[18]};
  const float* rb1[3] = {(const float*)d_in[11], (const float*)d_in[15], (const float*)d_in[19]};
  const float* rw2[3] = {(const float*)d_in[12], (const float*)d_in[16], (const float*)d_in[20]};
  const float* rb2[3] = {(const float*)d_in[13], (const float*)d_in[17], (const float*)d_in[21]};
  const float* start_out_w = (const float*)d_in[22];
  const float* start_out_b = (const float*)d_in[23];
  const float* term_w      = (const float*)d_in[24];
  const float* term_b      = (const float*)d_in[25];
  float* outp = (float*)d_out;

  // workspace layout (floats)
  float* base   = (float*)d_ws;
  const size_t MAT = (size_t)CST * HD;         // 2M floats
  float* bufA   = base;                        // embedding / residual input
  float* bufB   = base + MAT;                  // hidden H1
  float* xbuf   = base + 2 * MAT;              // sh (temporarily), then trans residual out
  float* nseb   = base + 3 * MAT;              // next embedding
  float* phb    = base + 4 * MAT;              // term residual out
  float* scores = base + 5 * MAT;              // 8192
  float* slse   = scores + CST;                // 1 (+pad)
  float* rowlse = slse + 16;                   // 8192
  float* Zb     = rowlse + CST;                // 8192
  float* emitb  = Zb + CST;                    // N*T*64
  int*   ib     = (int*)(emitb + (size_t)NBAT * TLEN * 64);
  int*   off    = ib;                          // 129
  int*   list   = ib + 132;                    // 10000

  const int gemmGrid = (CST / 16) * (HD / 16) / 8;       // 1024 blocks
  const int embGrid  = (int)(MAT / 256);                 // 8192 blocks
  const size_t DSH   = (size_t)64 * HD * sizeof(float);  // 64KB dyn LDS

  // ---- start branch first: sh lives in xbuf, consumed immediately (no aliasing) ----
  k_emb<<<embGrid, 256, 0, stream>>>(emb_c[0], emb_s[0], bufA);
  k_gemm<<<gemmGrid, 256, 0, stream>>>(bufA, rw1[0], rb1[0], nullptr, bufB, CST, HD, HD, 1);
  k_gemm<<<gemmGrid, 256, 0, stream>>>(bufB, rw2[0], rb2[0], bufA, xbuf, CST, HD, HD, 1);
  k_start_scores<<<CST / 64, 256, 0, stream>>>(xbuf, start_out_w, start_out_b, scores);
  k_vec_lse<<<1, 256, 0, stream>>>(scores, CST, slse);
  // ---- x = residual(emb(state), trans) -> xbuf (overwrites sh) ----
  k_emb<<<embGrid, 256, 0, stream>>>(emb_c[1], emb_s[1], bufA);
  k_gemm<<<gemmGrid, 256, 0, stream>>>(bufA, rw1[1], rb1[1], nullptr, bufB, CST, HD, HD, 1);
  k_gemm<<<gemmGrid, 256, 0, stream>>>(bufB, rw2[1], rb2[1], bufA, xbuf, CST, HD, HD, 1);
  // ---- nse = emb(next) ----
  k_emb<<<embGrid, 256, 0, stream>>>(emb_c[2], emb_s[2], nseb);
  // ---- ph = residual(emb(pre), term) ----
  k_emb<<<embGrid, 256, 0, stream>>>(emb_c[3], emb_s[3], bufA);
  k_gemm<<<gemmGrid, 256, 0, stream>>>(bufA, rw1[2], rb1[2], nullptr, bufB, CST, HD, HD, 1);
  k_gemm<<<gemmGrid, 256, 0, stream>>>(bufB, rw2[2], rb2[2], bufA, phb, CST, HD, HD, 1);
  // ---- transition row normalizers (streaming 8192x8192x256 online-LSE GEMM) ----
  k_row_lse<<<CST / 64, 256, DSH, stream>>>(xbuf, nseb, rowlse);
  // ---- deterministic word buckets per cluster ----
  k_offsets<<<1, KCL, 0, stream>>>(w2c, off);
  k_fill<<<KCL, 32, 0, stream>>>(w2c, off, list);
  // ---- emission normalizers and emit gather ----
  k_Z<<<KCL, 256, DSH, stream>>>(phb, term_w, term_b, off, list, Zb);
  k_emit<<<NBAT * TLEN, 256, 0, stream>>>(phb, term_w, term_b, Zb, text, w2c, emitb);
  // ---- forward recursion ----
  k_rec<<<NBAT, 256, 0, stream>>>(text, w2c, xbuf, nseb, rowlse, scores, slse, emitb, outp);
  (void)in_sizes; (void)n_in; (void)out_size; (void)ws_size;
}